// DConv_33629593928046
// MI455X (gfx1250) — compile-verified
//
#include <hip/hip_runtime.h>

typedef __attribute__((ext_vector_type(2))) float v2f;
typedef __attribute__((ext_vector_type(8))) float v8f;

#define B_    8
#define C_    256
#define H_    64
#define W_    64
#define HW_   (H_ * W_)
#define NOFF_ 18
#define KW_   (C_ * 9)   // 2304 = per-output-channel weight count

// ---------------------------------------------------------------------------
// Kernel 1: offsets = conv3x3(x, w_offset) via V_WMMA_F32_16X16X4_F32.
// One wave32 per 16-pixel row tile. Implicit GEMM, K ordered tap-major.
// Halo handling is branch-free: clamped address + 0/1 mask multiply, so all
// global loads are unconditional (no exec-mask save/restore in the hot loop).
//   A fragment (16x4 f32, 2 VGPRs): lanes 0-15 hold K=k0 (v0), k0+1 (v1);
//                                   lanes 16-31 hold K=k0+2, k0+3.
//   B fragment mirrors with N in lane[3:0].
//   D (16x16 f32, 8 VGPRs): VGPR r, lane l -> M = r + 8*(l>>4), N = l&15.
// Columns N>=18 of the second accumulator are never stored, so lanes m>=2
// load B row 0 unmasked and accumulate into dead columns.
// ---------------------------------------------------------------------------
__global__ __launch_bounds__(128) void dconv_offset_wmma(
    const float* __restrict__ x, const float* __restrict__ w,
    float* __restrict__ off)
{
  int wave = blockIdx.x * 4 + (threadIdx.x >> 5);   // 2048 tiles total
  int lane = threadIdx.x & 31;
  int xt = wave & 3;            // W / 16 = 4 tiles per row
  int y  = (wave >> 2) & 63;
  int b  = wave >> 8;
  int x0 = xt << 4;
  int m    = lane & 15;         // M row (pixel) / N col (weight)
  int half = lane >> 4;         // K sub-pair selector

  const float* xb = x + (size_t)b * C_ * HW_;

  v8f acc0 = {0.f, 0.f, 0.f, 0.f, 0.f, 0.f, 0.f, 0.f};
  v8f acc1 = acc0;

  int pxx = x0 + m;

  #pragma unroll
  for (int tap = 0; tap < 9; ++tap) {
    const int ki = tap / 3, kj = tap % 3;
    int yy = y + ki - 1;
    int xx = pxx + kj - 1;
    bool valid = (yy >= 0) && (yy < H_) && (xx >= 0) && (xx < W_);
    float am = valid ? 1.0f : 0.0f;           // A mask (data flow, no branch)
    int sidx = valid ? (yy * W_ + xx) : 0;    // clamped -> always in-bounds
    const float* ap = xb + sidx;              // + c*HW_ per channel
    const float* bp0 = w + (size_t)m * KW_ + tap;                 // N = m
    const float* bp1 = w + (size_t)(m < 2 ? (16 + m) : 0) * KW_ + tap;

    for (int c0 = 0; c0 < C_; c0 += 8) {
      int ca = c0 + (half << 1);
      int cb = ca + 4;
      v2f a0, a1, b00, b01, b10, b11;
      a0.x = ap[(size_t)ca * HW_] * am;
      a0.y = ap[(size_t)(ca + 1) * HW_] * am;
      a1.x = ap[(size_t)cb * HW_] * am;
      a1.y = ap[(size_t)(cb + 1) * HW_] * am;
      b00.x = bp0[ca * 9];
      b00.y = bp0[(ca + 1) * 9];
      b01.x = bp0[cb * 9];
      b01.y = bp0[(cb + 1) * 9];
      b10.x = bp1[ca * 9];
      b10.y = bp1[(ca + 1) * 9];
      b11.x = bp1[cb * 9];
      b11.y = bp1[(cb + 1) * 9];
      acc0 = __builtin_amdgcn_wmma_f32_16x16x4_f32(
          false, a0, false, b00, (short)0, acc0, false, false);
      acc1 = __builtin_amdgcn_wmma_f32_16x16x4_f32(
          false, a0, false, b10, (short)0, acc1, false, false);
      acc0 = __builtin_amdgcn_wmma_f32_16x16x4_f32(
          false, a1, false, b01, (short)0, acc0, false, false);
      acc1 = __builtin_amdgcn_wmma_f32_16x16x4_f32(
          false, a1, false, b11, (short)0, acc1, false, false);
    }
  }

  // Store D: off[b][n][y][x0+M]
  float* ob = off + (size_t)b * NOFF_ * HW_ + y * W_ + x0;
  #pragma unroll
  for (int r = 0; r < 8; ++r) {
    int M = r + (half << 3);
    ob[(size_t)m * HW_ + M] = acc0[r];
    if (m < 2) ob[(size_t)(16 + m) * HW_ + M] = acc1[r];
  }
}

// ---------------------------------------------------------------------------
// Kernel 2: deformable depthwise conv. Block = 256 threads handles one
// (b, y, 16-pixel) tile across all 256 channels. Geometry (bilinear weights
// + clamped gather indices) is channel-independent -> computed once into LDS
// and reused by all channels. Thread t: pixel = t&15 (coalesced stores),
// channel group cg = t>>4, channels c = cg + 16*cc. 64 independent gathers
// per tap per thread-iteration for memory-level parallelism; all gathers hit
// one batch slice (4.2 MB, L2-resident on 192 MB L2).
// ---------------------------------------------------------------------------
__global__ __launch_bounds__(256) void dconv_deform_dw(
    const float* __restrict__ x, const float* __restrict__ off,
    const float* __restrict__ wd, float* __restrict__ out)
{
  __shared__ float s_off[NOFF_ * 16];   // [off-channel][pixel]
  __shared__ float s_w[C_ * 9];         // depthwise weights [c][tap]
  __shared__ float s_gw[144 * 4];       // (px,tap) -> 4 corner weights
  __shared__ int   s_gi[144 * 4];       // (px,tap) -> 4 clamped flat indices

  int tid = threadIdx.x;
  int bid = blockIdx.x;                 // 2048 tiles
  int xt = bid & 3;
  int y  = (bid >> 2) & 63;
  int b  = bid >> 8;
  int x0 = xt << 4;

  const float* xb = x + (size_t)b * C_ * HW_;

  // Warm L2/near caches for the rows this tile will gather from
  // (emits global_prefetch_b8 on gfx1250).
  {
    const float* prow = xb + (size_t)(tid >> 4) * 16 * HW_ + y * W_;
    __builtin_prefetch(prow, 0, 1);
  }

  // Stage offsets for this tile (coalesced over pixels).
  const float* offb = off + (size_t)b * NOFF_ * HW_ + y * W_ + x0;
  for (int idx = tid; idx < NOFF_ * 16; idx += 256) {
    int ch = idx >> 4, p = idx & 15;
    s_off[ch * 16 + p] = offb[(size_t)ch * HW_ + p];
  }
  // Stage all depthwise weights.
  for (int idx = tid; idx < C_ * 9; idx += 256) s_w[idx] = wd[idx];
  __syncthreads();

  // Precompute bilinear geometry per (pixel, tap): 144 entries.
  for (int e = tid; e < 144; e += 256) {
    int p = e / 9, tap = e - p * 9;
    int ky = tap / 3, kx = tap - ky * 3;
    float py  = s_off[(2 * tap) * 16 + p]     + (float)(y + ky - 1);
    float pxf = s_off[(2 * tap + 1) * 16 + p] + (float)(x0 + p + kx - 1);
    float fy0 = floorf(py), fx0 = floorf(pxf);
    int iy0 = (int)fy0, ix0 = (int)fx0;
    int iy1 = iy0 + 1,  ix1 = ix0 + 1;
    float ty = py - fy0, tx = pxf - fx0;
    float wy0 = (1.f - ty) * ((iy0 >= 0 && iy0 < H_) ? 1.f : 0.f);
    float wy1 = ty         * ((iy1 >= 0 && iy1 < H_) ? 1.f : 0.f);
    float wx0 = (1.f - tx) * ((ix0 >= 0 && ix0 < W_) ? 1.f : 0.f);
    float wx1 = tx         * ((ix1 >= 0 && ix1 < W_) ? 1.f : 0.f);
    int iy0c = min(max(iy0, 0), H_ - 1), iy1c = min(max(iy1, 0), H_ - 1);
    int ix0c = min(max(ix0, 0), W_ - 1), ix1c = min(max(ix1, 0), W_ - 1);
    s_gw[e * 4 + 0] = wy0 * wx0;
    s_gw[e * 4 + 1] = wy0 * wx1;
    s_gw[e * 4 + 2] = wy1 * wx0;
    s_gw[e * 4 + 3] = wy1 * wx1;
    s_gi[e * 4 + 0] = iy0c * W_ + ix0c;
    s_gi[e * 4 + 1] = iy0c * W_ + ix1c;
    s_gi[e * 4 + 2] = iy1c * W_ + ix0c;
    s_gi[e * 4 + 3] = iy1c * W_ + ix1c;
  }
  __syncthreads();

  int p  = tid & 15;
  int cg = tid >> 4;

  float acc[16];
  #pragma unroll
  for (int cc = 0; cc < 16; ++cc) acc[cc] = 0.f;

  for (int tap = 0; tap < 9; ++tap) {
    int g = (p * 9 + tap) * 4;
    float w00 = s_gw[g + 0], w01 = s_gw[g + 1];
    float w10 = s_gw[g + 2], w11 = s_gw[g + 3];
    int   i00 = s_gi[g + 0], i01 = s_gi[g + 1];
    int   i10 = s_gi[g + 2], i11 = s_gi[g + 3];
    #pragma unroll
    for (int cc = 0; cc < 16; ++cc) {
      int c = cg + (cc << 4);
      const float* xc = xb + (size_t)c * HW_;
      float v = w00 * xc[i00] + w01 * xc[i01]
              + w10 * xc[i10] + w11 * xc[i11];
      acc[cc] += s_w[c * 9 + tap] * v;
    }
  }

  float* ob = out + (size_t)b * C_ * HW_ + y * W_ + x0 + p;
  #pragma unroll
  for (int cc = 0; cc < 16; ++cc) {
    int c = cg + (cc << 4);
    ob[(size_t)c * HW_] = acc[cc];
  }
}

// ---------------------------------------------------------------------------
extern "C" void kernel_launch(void* const* d_in, const int* in_sizes, int n_in,
                              void* d_out, int out_size, void* d_ws,
                              size_t ws_size, hipStream_t stream) {
  const float* x        = (const float*)d_in[0];   // [8,256,64,64]
  const float* w_offset = (const float*)d_in[1];   // [18,256,3,3]
  const float* w_deform = (const float*)d_in[2];   // [256,1,3,3]
  float* out = (float*)d_out;                      // [8,256,64,64]
  float* off = (float*)d_ws;                       // [8,18,64,64] = 2.36 MB

  // 2048 wave-tiles, 4 waves per block.
  dconv_offset_wmma<<<512, 128, 0, stream>>>(x, w_offset, off);
  // 2048 pixel tiles x 256 channels.
  dconv_deform_dw<<<2048, 256, 0, stream>>>(x, off, w_deform, out);
}